// CARAFEPack_9680856285232
// MI455X (gfx1250) — compile-verified
//
#include <hip/hip_runtime.h>
#include <hip/hip_bf16.h>

// CARAFE upsample (K=5, S=2): 1x1 conv -> 3x3 conv -> pixel_shuffle+softmax -> reassembly.
// fp32 WMMA (V_WMMA_F32_16X16X4_F32) for both convs (codegen-verified in round 1);
// reassembly is the memory-bound stage: double-buffered LDS staging + non-temporal
// streaming stores so the 67 MB output doesn't evict L2-resident x / weights.

#define NB     4
#define CIN    256
#define CCOMP  64
#define CMASK  100      // K*K*S*S
#define HH     64
#define WW     64
#define HW     (HH*WW)          // 4096
#define HO     128
#define WO     128
#define HWO    (HO*WO)          // 16384
#define KTOT   576              // CCOMP * 3 * 3
#define MPAD   112              // CMASK padded to 7 tiles of 16

typedef float v2f __attribute__((ext_vector_type(2)));
typedef float v8f __attribute__((ext_vector_type(8)));

// ---------------------------------------------------------------------------
// 1) Repack encoder weights: [100][64][3][3] -> [112][plane=dy*3+dx][ci], zero pad M.
//    GEMM K index k' = plane*64 + ci (contiguous ci per tap -> trivial B addressing).
// ---------------------------------------------------------------------------
__global__ void repack_wenc(const float* __restrict__ w_enc,
                            float* __restrict__ w_encT) {
    int idx = blockIdx.x * 256 + threadIdx.x;
    if (idx >= MPAD * KTOT) return;
    int m     = idx / KTOT;
    int r     = idx % KTOT;
    int plane = r >> 6;       // dy*3+dx
    int ci    = r & 63;
    float v = 0.0f;
    if (m < CMASK) v = w_enc[m * KTOT + ci * 9 + plane];
    w_encT[idx] = v;
}

// ---------------------------------------------------------------------------
// 2) 1x1 conv as GEMM: comp[n][co][hw] = sum_ci w_comp[co][ci] * x[n][ci][hw] + b.
//    One wave handles a 16-wide column strip and ALL of M (4 tiles, 4 accumulators)
//    so x (B matrix) is loaded exactly once. K loop: 64 steps of 4 -> 256 WMMAs/wave.
//    A-frag: lane holds A[m=lane%16][k = kb + 2*(lane/16) + {0,1}]  (v2f, 8B aligned)
//    B-frag: lane holds B[k][n=lane%16] same K split.
// ---------------------------------------------------------------------------
__global__ void __launch_bounds__(128)
compress_gemm(const float* __restrict__ x, const float* __restrict__ w_comp,
              const float* __restrict__ b_comp, float* __restrict__ comp) {
    const int wid   = blockIdx.x * 4 + (threadIdx.x >> 5);
    const int lane  = threadIdx.x & 31;
    const int n     = wid >> 8;            // 0..3
    const int nt    = wid & 255;           // 256 column tiles of 16 over HW=4096
    const int col   = nt * 16 + (lane & 15);
    const int khalf = (lane >> 4) * 2;     // 0 or 2
    const int mrow  = lane & 15;

    const float* xb = x + (size_t)n * CIN * HW + col;

    v8f acc[4];
    const v8f vzero = {0.f,0.f,0.f,0.f,0.f,0.f,0.f,0.f};
#pragma unroll
    for (int mt = 0; mt < 4; ++mt) acc[mt] = vzero;

    for (int kb = 0; kb < CIN; kb += 4) {
        const int k = kb + khalf;
        v2f bfrag;
        bfrag.x = xb[(size_t)k * HW];
        bfrag.y = xb[(size_t)(k + 1) * HW];
#pragma unroll
        for (int mt = 0; mt < 4; ++mt) {
            v2f afrag = *(const v2f*)(w_comp + (size_t)(mt * 16 + mrow) * CIN + k);
            acc[mt] = __builtin_amdgcn_wmma_f32_16x16x4_f32(
                false, afrag, false, bfrag, (short)0, acc[mt], false, false);
        }
    }

    // C/D layout: lane l, vgpr r -> (m = r + 8*(l/16), ncol = l%16)
    float* cb = comp + (size_t)n * CCOMP * HW + col;
    const int mhi = 8 * (lane >> 4);
#pragma unroll
    for (int mt = 0; mt < 4; ++mt) {
#pragma unroll
        for (int r = 0; r < 8; ++r) {
            const int m = mt * 16 + r + mhi;
            cb[(size_t)m * HW] = acc[mt][r] + b_comp[m];
        }
    }
}

// ---------------------------------------------------------------------------
// 3) 3x3 conv as implicit GEMM: M=112(pad of 100), K=576, N=4096 per image.
//    One wave per 16-pixel strip, 7 M-tile accumulators (B-frags built once per
//    k-step, reused by 7 WMMAs). Outer loop over 9 taps; halo handled branch-free
//    (clamped address + select) so EXEC stays all-ones for WMMA.
// ---------------------------------------------------------------------------
__global__ void __launch_bounds__(128)
encoder_gemm(const float* __restrict__ comp, const float* __restrict__ w_encT,
             const float* __restrict__ b_enc, float* __restrict__ mask) {
    const int wid   = blockIdx.x * 4 + (threadIdx.x >> 5);
    const int lane  = threadIdx.x & 31;
    const int n     = wid >> 8;
    const int nt    = wid & 255;
    const int col   = nt * 16 + (lane & 15);
    const int h     = col >> 6;
    const int w     = col & 63;
    const int khalf = (lane >> 4) * 2;
    const int mrow  = lane & 15;

    v8f acc[7];
    const v8f vzero = {0.f,0.f,0.f,0.f,0.f,0.f,0.f,0.f};
#pragma unroll
    for (int mt = 0; mt < 7; ++mt) acc[mt] = vzero;

    const float* cbase = comp + (size_t)n * CCOMP * HW;

#pragma unroll
    for (int plane = 0; plane < 9; ++plane) {
        const int dy = plane / 3, dx = plane % 3;
        const int y  = h + dy - 1;
        const int xw = w + dx - 1;
        const bool valid = (y >= 0) && (y < HH) && (xw >= 0) && (xw < WW);
        const int yc = min(max(y, 0), HH - 1);
        const int xc = min(max(xw, 0), WW - 1);
        const float* bsrc = cbase + yc * WW + xc;

#pragma unroll 4
        for (int cib = 0; cib < CCOMP; cib += 4) {
            const int ci = cib + khalf;
            float b0 = bsrc[(size_t)ci * HW];
            float b1 = bsrc[(size_t)(ci + 1) * HW];
            v2f bfrag;
            bfrag.x = valid ? b0 : 0.0f;
            bfrag.y = valid ? b1 : 0.0f;
            const int kk = plane * 64 + ci;   // even -> 8B-aligned v2f loads
#pragma unroll
            for (int mt = 0; mt < 7; ++mt) {
                v2f afrag = *(const v2f*)(w_encT + (size_t)(mt * 16 + mrow) * KTOT + kk);
                acc[mt] = __builtin_amdgcn_wmma_f32_16x16x4_f32(
                    false, afrag, false, bfrag, (short)0, acc[mt], false, false);
            }
        }
    }

    float* mbase = mask + (size_t)n * CMASK * HW + col;
    const int mhi = 8 * (lane >> 4);
#pragma unroll
    for (int mt = 0; mt < 7; ++mt) {
#pragma unroll
        for (int r = 0; r < 8; ++r) {
            const int cm = mt * 16 + r + mhi;
            if (cm < CMASK) mbase[(size_t)cm * HW] = acc[mt][r] + b_enc[cm];
        }
    }
}

// ---------------------------------------------------------------------------
// 4) pixel_shuffle + softmax over the 25 taps.
//    m[n][k][2h+sh][2w+sw] = softmax_k( mask[n][k*4 + sh*2 + sw][h][w] )
// ---------------------------------------------------------------------------
__global__ void softmax_shuffle(const float* __restrict__ mask,
                                float* __restrict__ msoft) {
    int t = blockIdx.x * 256 + threadIdx.x;
    if (t >= NB * HWO) return;
    int n   = t / HWO;
    int rem = t % HWO;
    int Hr  = rem / WO;
    int Wj  = rem % WO;
    int h = Hr >> 1, sh = Hr & 1;
    int w = Wj >> 1, sw = Wj & 1;
    int r = sh * 2 + sw;

    const float* mb = mask + (size_t)n * CMASK * HW + h * WW + w;
    float v[25];
    float mx = -1e30f;
#pragma unroll
    for (int k = 0; k < 25; ++k) {
        v[k] = mb[(size_t)(k * 4 + r) * HW];
        mx = fmaxf(mx, v[k]);
    }
    float s = 0.0f;
#pragma unroll
    for (int k = 0; k < 25; ++k) { v[k] = __expf(v[k] - mx); s += v[k]; }
    float inv = 1.0f / s;
    float* ob = msoft + (size_t)n * 25 * HWO + Hr * WO + Wj;
#pragma unroll
    for (int k = 0; k < 25; ++k) ob[(size_t)k * HWO] = v[k] * inv;
}

// ---------------------------------------------------------------------------
// 5) Reassembly: the HBM-bound stage (67 MB streaming out of ~100 MB total).
//    Block per (n, low-res row h); thread owns one hi-res pixel of the row-pair.
//    - 25 softmax weights in registers (single-use -> non-temporal loads)
//    - per-channel 5x68 x-halo tile double-buffered in LDS: c+1 global loads
//      are issued before computing channel c; ONE barrier per channel.
//    - output written with non-temporal stores (write-once stream, keep L2
//      for x which is re-read by 5 neighboring row-blocks).
// ---------------------------------------------------------------------------
__global__ void __launch_bounds__(256)
reassemble(const float* __restrict__ x, const float* __restrict__ msoft,
           float* __restrict__ out) {
    const int n   = blockIdx.x >> 6;
    const int h   = blockIdx.x & 63;
    const int t   = threadIdx.x;
    const int sh  = t >> 7;          // 0/1
    const int j   = t & 127;         // hi-res column
    const int wlo = j >> 1;          // low-res column
    const int Hr  = 2 * h + sh;

    float wt[25];
    const float* mb = msoft + (size_t)n * 25 * HWO + Hr * WO + j;
#pragma unroll
    for (int k = 0; k < 25; ++k)
        wt[k] = __builtin_nontemporal_load(&mb[(size_t)k * HWO]);

    // Per-thread staging slots for the 5x68 = 340-element halo tile.
    const int  idx0 = t;
    const int  row0 = idx0 / 68, col0 = idx0 % 68;
    const int  yr0  = h + row0 - 2, xc0 = col0 - 2;
    const bool v0   = (yr0 >= 0) && (yr0 < HH) && (xc0 >= 0) && (xc0 < WW);
    const size_t off0 = v0 ? (size_t)(yr0 * WW + xc0) : 0;

    const int  idx1 = t + 256;
    const bool has1 = idx1 < 5 * 68;
    const int  row1 = idx1 / 68, col1 = idx1 % 68;
    const int  yr1  = h + row1 - 2, xc1 = col1 - 2;
    const bool v1   = has1 && (yr1 >= 0) && (yr1 < HH) && (xc1 >= 0) && (xc1 < WW);
    const size_t off1 = v1 ? (size_t)(yr1 * WW + xc1) : 0;

    __shared__ float tile[2][5 * 68];
    const float* xb = x + (size_t)n * CIN * HW;
    float* ob = out + (size_t)n * CIN * HWO + Hr * WO + j;

    // Prologue: stage channel 0 into buffer 0.
    {
        float s0 = v0 ? xb[off0] : 0.0f;
        float s1 = v1 ? xb[off1] : 0.0f;
        tile[0][idx0] = s0;
        if (has1) tile[0][idx1] = s1;
    }
    __syncthreads();

    for (int c = 0; c < CIN; ++c) {
        const int cur = c & 1;
        const int nxt = cur ^ 1;

        // Issue next channel's global loads early (latency hidden by compute).
        float s0 = 0.0f, s1 = 0.0f;
        if (c + 1 < CIN) {
            const float* xc = xb + (size_t)(c + 1) * HW;
            s0 = v0 ? xc[off0] : 0.0f;
            s1 = v1 ? xc[off1] : 0.0f;
        }

        float acc = 0.0f;
#pragma unroll
        for (int p = 0; p < 5; ++p)
#pragma unroll
            for (int q = 0; q < 5; ++q)
                acc = fmaf(wt[p * 5 + q], tile[cur][p * 68 + wlo + q], acc);

        __builtin_nontemporal_store(acc, &ob[(size_t)c * HWO]);

        if (c + 1 < CIN) {
            tile[nxt][idx0] = s0;
            if (has1) tile[nxt][idx1] = s1;
        }
        __syncthreads();   // reads of tile[cur] & writes of tile[nxt] both fenced
    }
}

// ---------------------------------------------------------------------------
extern "C" void kernel_launch(void* const* d_in, const int* in_sizes, int n_in,
                              void* d_out, int out_size, void* d_ws, size_t ws_size,
                              hipStream_t stream) {
    const float* x      = (const float*)d_in[0];
    const float* w_comp = (const float*)d_in[1];
    const float* b_comp = (const float*)d_in[2];
    const float* w_enc  = (const float*)d_in[3];
    const float* b_enc  = (const float*)d_in[4];
    float* out = (float*)d_out;

    // Workspace layout (bytes):
    //   comp   : 4*64*4096*4   = 4,194,304
    //   mask   : 4*100*4096*4  = 6,553,600
    //   msoft  : 4*25*16384*4  = 6,553,600
    //   w_encT : 112*576*4     =   258,048     (total ~17.6 MB)
    char* ws = (char*)d_ws;
    float* comp   = (float*)(ws);
    float* mask   = (float*)(ws + (size_t)4194304);
    float* msoft  = (float*)(ws + (size_t)4194304 + 6553600);
    float* w_encT = (float*)(ws + (size_t)4194304 + 6553600 + 6553600);

    repack_wenc    <<<(MPAD * KTOT + 255) / 256, 256, 0, stream>>>(w_enc, w_encT);
    compress_gemm  <<<256, 128, 0, stream>>>(x, w_comp, b_comp, comp);
    encoder_gemm   <<<256, 128, 0, stream>>>(comp, w_encT, b_enc, mask);
    softmax_shuffle<<<(NB * HWO + 255) / 256, 256, 0, stream>>>(mask, msoft);
    reassemble     <<<NB * 64, 256, 0, stream>>>(x, msoft, out);
}